// INB_35201551958512
// MI455X (gfx1250) — compile-verified
//
#include <hip/hip_runtime.h>

typedef float v2f __attribute__((ext_vector_type(2)));
typedef float v8f __attribute__((ext_vector_type(8)));

#define N_ROWS   131072
#define DIM      256
#define KP       16
#define NDOM     8
#define NBIN     32
#define NEDGE    33
#define GQ       100

#define WAVES_PER_BLOCK 4
#define TPB             (WAVES_PER_BLOCK * 32)
#define ROWS_PER_WAVE   16
#define ROWS_PER_BLOCK  (WAVES_PER_BLOCK * ROWS_PER_WAVE)

// LDS pitches chosen for bank-conflict freedom AND 8-byte alignment of b64 frags
#define WTT_PITCH 258   // even; 258 mod 64 = 2 -> rows 0..15 hit distinct bank pairs
#define DT_PITCH  18    // even; 18  mod 64 -> rows 0..15 distinct bank pairs

static __device__ __forceinline__ v8f wmma4(v2f a, v2f b, v8f c) {
  // V_WMMA_F32_16X16X4_F32 : D(16x16,f32) = A(16x4,f32) * B(4x16,f32) + C
  return __builtin_amdgcn_wmma_f32_16x16x4_f32(
      /*neg_a=*/false, a, /*neg_b=*/false, b,
      /*c_mod=*/(short)0, c, /*reuse_a=*/false, /*reuse_b=*/false);
}

__launch_bounds__(TPB, 1)
__global__ void inb_fused_kernel(const float* __restrict__ X,
                                 const int*   __restrict__ y,
                                 const float* __restrict__ wT,
                                 const float* __restrict__ bin_edges,
                                 const float* __restrict__ cdf_vals,
                                 const float* __restrict__ bary_q,
                                 float*       __restrict__ out)
{
  __shared__ float s_wTt[KP * WTT_PITCH];                       // wTt[k][d] transposed
  __shared__ float s_edges[NDOM * KP * NEDGE];
  __shared__ float s_cdf[NDOM * KP * NEDGE];
  __shared__ float s_bq[KP * GQ];
  __shared__ float s_dt[WAVES_PER_BLOCK * ROWS_PER_WAVE * DT_PITCH];
  __shared__ int   s_y[ROWS_PER_BLOCK];

  const int tid = threadIdx.x;
  const int rowbase_blk = blockIdx.x * ROWS_PER_BLOCK;

  // ---- stage tables into LDS (all tiny; L2-resident in global) ----
  for (int i = tid; i < DIM * KP; i += TPB) {
    const int d = i >> 4;           // wT is (256,16) row-major
    const int k = i & 15;
    s_wTt[k * WTT_PITCH + d] = wT[i];
  }
  for (int i = tid; i < NDOM * KP * NEDGE; i += TPB) {
    s_edges[i] = bin_edges[i];
    s_cdf[i]   = cdf_vals[i];
  }
  for (int i = tid; i < KP * GQ; i += TPB) s_bq[i] = bary_q[i];
  if (tid < ROWS_PER_BLOCK) s_y[tid] = y[rowbase_blk + tid];
  __syncthreads();

  const int wave = tid >> 5;
  const int lane = tid & 31;
  const int lh   = lane >> 4;       // half-wave select (ISA frag layouts)
  const int lm   = lane & 15;
  const int rowbase = rowbase_blk + wave * ROWS_PER_WAVE;

  // =========== Phase 1: Xm tile (16 rows x 16 k) = X_tile @ wT ===========
  // A frag (16x4 f32): lane<16 -> M=lane, K=0,1 ; lane>=16 -> M=lane-16, K=2,3
  const float* aptr = X + (size_t)(rowbase + lm) * DIM + 2 * lh;
  // B frag (4x16 f32): vgpr0 = rows {0|2}, vgpr1 = rows {1|3}; contiguous in wTt[k][d]
  const float* bptr = &s_wTt[lm * WTT_PITCH + 2 * lh];

  v8f acc0 = {0.f,0.f,0.f,0.f,0.f,0.f,0.f,0.f};
  v8f acc1 = {0.f,0.f,0.f,0.f,0.f,0.f,0.f,0.f};
#pragma unroll 8
  for (int kc = 0; kc < 64; kc += 2) {
    v2f a0 = *(const v2f*)(aptr + 4 * kc);
    v2f b0 = *(const v2f*)(bptr + 4 * kc);
    v2f a1 = *(const v2f*)(aptr + 4 * kc + 4);
    v2f b1 = *(const v2f*)(bptr + 4 * kc + 4);
    acc0 = wmma4(a0, b0, acc0);
    acc1 = wmma4(a1, b1, acc1);
  }
  v8f xm = acc0 + acc1;   // C layout: elem j -> (M = j + 8*lh, N = lm)

  // =========== Phase 2: histogram CDF + barycenter inverse-CDF ===========
  float dcf[8];
#pragma unroll
  for (int j = 0; j < 8; ++j) {
    const float v    = xm[j];
    const int   rloc = wave * ROWS_PER_WAVE + j + 8 * lh;   // row within block
    const int   dom  = s_y[rloc];
    const float* e = &s_edges[(dom * KP + lm) * NEDGE];
    const float* c = &s_cdf [(dom * KP + lm) * NEDGE];
    // branchless binary search: largest bi in [0,31] with e[bi] <= v (clipped)
    int bi = (e[16] <= v) ? 16 : 0;
    bi = (e[bi + 8] <= v) ? bi + 8 : bi;
    bi = (e[bi + 4] <= v) ? bi + 4 : bi;
    bi = (e[bi + 2] <= v) ? bi + 2 : bi;
    bi = (e[bi + 1] <= v) ? bi + 1 : bi;
    const float x0 = e[bi], x1 = e[bi + 1];
    const float f0 = c[bi], f1 = c[bi + 1];
    float t = (v - x0) / (x1 - x0);
    t = fminf(fmaxf(t, 0.f), 1.f);
    float u = f0 + t * (f1 - f0);
    u = fminf(fmaxf(u, 0.f), 1.f);
    const float pos = u * (float)(GQ - 1);
    int jj = (int)pos;
    jj = jj < 0 ? 0 : (jj > GQ - 2 ? GQ - 2 : jj);
    const float tt = pos - (float)jj;
    const float* bq = &s_bq[lm * GQ];
    const float z = bq[jj] * (1.f - tt) + bq[jj + 1] * tt;
    dcf[j] = z - v;                 // fused: out = X + (z - Xm) @ wT^T
  }

  // relayout D tile through LDS (C layout -> A-fragment layout)
  {
    float* dt = &s_dt[wave * ROWS_PER_WAVE * DT_PITCH];
#pragma unroll
    for (int j = 0; j < 8; ++j)
      dt[(j + 8 * lh) * DT_PITCH + lm] = dcf[j];
  }
  __syncthreads();   // order the cross-lane LDS store->load

  // =========== Phase 3: out_tile = X_tile + D_tile(16x16) @ wT^T ===========
  const float* arow2 = &s_dt[wave * ROWS_PER_WAVE * DT_PITCH + lm * DT_PITCH + 2 * lh];
#pragma unroll
  for (int nc = 0; nc < 16; ++nc) {
    const int colbase = nc * 16 + lm;
    v8f cacc;
#pragma unroll
    for (int j = 0; j < 8; ++j)   // init C with X (coalesced 64B segments)
      cacc[j] = X[(size_t)(rowbase + j + 8 * lh) * DIM + colbase];
#pragma unroll
    for (int kc = 0; kc < 4; ++kc) {
      v2f a = *(const v2f*)(arow2 + 4 * kc);
      // B[kk][n] = wT[col][kbase+kk] = wTt[kbase+kk][col]
      v2f b = { s_wTt[(kc * 4 + 2 * lh + 0) * WTT_PITCH + colbase],
                s_wTt[(kc * 4 + 2 * lh + 1) * WTT_PITCH + colbase] };
      cacc = wmma4(a, b, cacc);
    }
#pragma unroll
    for (int j = 0; j < 8; ++j)   // streaming output: non-temporal stores
      __builtin_nontemporal_store(
          cacc[j], &out[(size_t)(rowbase + j + 8 * lh) * DIM + colbase]);
  }
}

extern "C" void kernel_launch(void* const* d_in, const int* in_sizes, int n_in,
                              void* d_out, int out_size, void* d_ws, size_t ws_size,
                              hipStream_t stream) {
  (void)in_sizes; (void)n_in; (void)out_size; (void)d_ws; (void)ws_size;
  const float* X  = (const float*)d_in[0];
  const int*   y  = (const int*)  d_in[1];
  const float* wT = (const float*)d_in[2];
  const float* be = (const float*)d_in[3];
  const float* cv = (const float*)d_in[4];
  const float* bq = (const float*)d_in[5];
  float* out = (float*)d_out;

  dim3 grid(N_ROWS / ROWS_PER_BLOCK);   // 2048 blocks
  dim3 block(TPB);                      // 128 threads = 4 wave32
  hipLaunchKernelGGL(inb_fused_kernel, grid, block, 0, stream,
                     X, y, wT, be, cv, bq, out);
}